// PooledCausalLMLoss_8375186227911
// MI455X (gfx1250) — compile-verified
//
#include <hip/hip_runtime.h>
#include <hip/hip_bf16.h>
#include <math.h>

// Problem constants (from reference): B=4, T=2048, V=32000, S=17
#define BB 4
#define TT 2048
#define VV 32000
#define SS 17
#define ROWS (BB * (TT - 1))        // 8188 softmax rows
#define VEC4 (VV / 4)               // 8000 float4 per row
#define THREADS 256
#define CHUNK_V4 1024               // float4 per chunk (4096 floats, 16 KB)
#define NCHUNK 8                    // ceil(8000/1024)
#define QPT 4                       // vec4 issues per thread per chunk (1024/256)
#define NBUF 3                      // triple buffer: 2 chunks in flight (Little's law)

#define LOG2E 1.4426950408889634f
#define LN2   0.6931471805599453f

// ---- single-instruction exp2/log2 (v_exp_f32 / v_log_f32) --------------------------
#if __has_builtin(__builtin_amdgcn_exp2f)
__device__ __forceinline__ float fast_exp2(float x) { return __builtin_amdgcn_exp2f(x); }
#else
__device__ __forceinline__ float fast_exp2(float x) { return __expf(x * LN2); }
#endif
#if __has_builtin(__builtin_amdgcn_logf)
__device__ __forceinline__ float fast_log2(float x) { return __builtin_amdgcn_logf(x); }
#else
__device__ __forceinline__ float fast_log2(float x) { return __logf(x) * LOG2E; }
#endif

// ---- CDNA5 async global->LDS data mover (inline asm; portable across toolchains) ----
typedef __attribute__((address_space(3))) void lds_void;

__device__ __forceinline__ unsigned lds_off(const void* p) {
  // generic (flat) -> LDS segment pointer -> 32-bit LDS byte offset
  return (unsigned)(unsigned long long)(lds_void*)p;
}

__device__ __forceinline__ void async_load_b128(unsigned lds_byte_off, const void* gaddr) {
  // GLOBAL_LOAD_ASYNC_TO_LDS_B128: VDST = LDS byte address VGPR, VADDR = 64-bit global addr
  asm volatile("global_load_async_to_lds_b128 %0, %1, off"
               :: "v"(lds_byte_off), "v"(gaddr)
               : "memory");
}

template <int N>
__device__ __forceinline__ void wait_asynccnt() {
  asm volatile("s_wait_asynccnt %0" :: "n"(N) : "memory");
}

// Constant-folded dispatcher (asm "n" needs a parse-time constant; after full
// unroll the selector is a compile-time constant and the switch folds away).
__device__ __forceinline__ void wait_async_chunks(int chunks_outstanding) {
  switch (chunks_outstanding) {
    case 2:  wait_asynccnt<2 * QPT>(); break;
    case 1:  wait_asynccnt<1 * QPT>(); break;
    default: wait_asynccnt<0>();       break;
  }
}

// Merge two (max, sumexp) pairs.
__device__ __forceinline__ void lse_merge(float& m, float& s, float mo, float so) {
  float M = fmaxf(m, mo);
  s = s * fast_exp2((m - M) * LOG2E) + so * fast_exp2((mo - M) * LOG2E);
  m = M;
}

// -------------------------------------------------------------------------------------
__global__ void zero_ws_kernel(float* ws, int n) {
  int i = blockIdx.x * blockDim.x + threadIdx.x;
  if (i < n) ws[i] = 0.0f;
}

// One workgroup (8 waves) per softmax row. Streams the row through LDS with
// triple-buffered async copies (2 chunks / 4 KB per wave in flight); each
// thread consumes only the LDS slots it filled, so only per-wave
// s_wait_asynccnt ordering is required — no barriers in the steady state.
// Per chunk each thread does a branch-free two-phase (max tree, then
// fma+exp2+add per element) — ~3 VALU + 1 trans per element.
__global__ void __launch_bounds__(THREADS)
row_ce_kernel(const float* __restrict__ logits,
              const int* __restrict__ input_ids,
              const int* __restrict__ loss_mask,
              const int* __restrict__ step_indices,
              float* __restrict__ sums,
              float* __restrict__ cnts) {
  extern __shared__ float4 lds4[];            // NBUF * CHUNK_V4 float4 = 48 KB
  const int row = blockIdx.x;                 // 0..ROWS-1
  const int b = row / (TT - 1);
  const int t = row % (TT - 1);
  const float* rowp = logits + ((size_t)b * TT + t) * VV;
  const float4* row4 = (const float4*)rowp;
  const int tid = threadIdx.x;

  // Issue one chunk's worth of async b128 copies (uniform QPT issues/thread).
  auto issue_chunk = [&](int c, int buf) {
#pragma unroll
    for (int q = 0; q < QPT; ++q) {
      int p = tid + q * THREADS;              // slot within chunk (< CHUNK_V4)
      int v = c * CHUNK_V4 + p;
      if (v > VEC4 - 1) v = VEC4 - 1;         // clamp tail (keeps issue count uniform)
      async_load_b128(lds_off(&lds4[buf * CHUNK_V4 + p]), row4 + v);
    }
  };

  // Prefetch chunks 0 and 1 (pipeline depth 2).
  issue_chunk(0, 0);
  issue_chunk(1, 1);

  float m = -INFINITY, s = 0.0f;

#pragma unroll
  for (int c = 0; c < NCHUNK; ++c) {
    const int buf = c % NBUF;
    if (c + 2 < NCHUNK) issue_chunk(c + 2, (c + 2) % NBUF);
    const int rem = NCHUNK - 1 - c;           // chunks still outstanding after chunk c
    wait_async_chunks(rem > 2 ? 2 : rem);     // chunk c has landed (in-order per wave)

    // ---- load this thread's 16 elements of the chunk from LDS (ds_load_b128) ----
    float4 x0 = lds4[buf * CHUNK_V4 + tid + 0 * THREADS];
    float4 x1 = lds4[buf * CHUNK_V4 + tid + 1 * THREADS];
    float4 x2 = lds4[buf * CHUNK_V4 + tid + 2 * THREADS];
    float4 x3 = lds4[buf * CHUNK_V4 + tid + 3 * THREADS];

    // Sanitize tail vec4s (only the last chunk can be out of range): -inf
    // contributes 0 to the sum and does not move the max.
    const int vb = c * CHUNK_V4 + tid;
    if (vb + 1 * THREADS >= VEC4) x1.x = x1.y = x1.z = x1.w = -INFINITY;
    if (vb + 2 * THREADS >= VEC4) x2.x = x2.y = x2.z = x2.w = -INFINITY;
    if (vb + 3 * THREADS >= VEC4) x3.x = x3.y = x3.z = x3.w = -INFINITY;

    // ---- phase 1: max tree over 16 values ----
    float a0 = fmaxf(fmaxf(x0.x, x0.y), fmaxf(x0.z, x0.w));
    float a1 = fmaxf(fmaxf(x1.x, x1.y), fmaxf(x1.z, x1.w));
    float a2 = fmaxf(fmaxf(x2.x, x2.y), fmaxf(x2.z, x2.w));
    float a3 = fmaxf(fmaxf(x3.x, x3.y), fmaxf(x3.z, x3.w));
    float m16 = fmaxf(fmaxf(a0, a1), fmaxf(a2, a3));
    float mnew = fmaxf(m, m16);
    float mnl = mnew * LOG2E;

    // rescale running sum (exp2(-inf) = 0 handles the first chunk)
    s = s * fast_exp2(__fmaf_rn(m, LOG2E, -mnl));
    m = mnew;

    // ---- phase 2: one fma + exp2 + add per element, 4 independent chains ----
    float p0 = fast_exp2(__fmaf_rn(x0.x, LOG2E, -mnl));
    float p1 = fast_exp2(__fmaf_rn(x0.y, LOG2E, -mnl));
    float p2 = fast_exp2(__fmaf_rn(x0.z, LOG2E, -mnl));
    float p3 = fast_exp2(__fmaf_rn(x0.w, LOG2E, -mnl));
    p0 += fast_exp2(__fmaf_rn(x1.x, LOG2E, -mnl));
    p1 += fast_exp2(__fmaf_rn(x1.y, LOG2E, -mnl));
    p2 += fast_exp2(__fmaf_rn(x1.z, LOG2E, -mnl));
    p3 += fast_exp2(__fmaf_rn(x1.w, LOG2E, -mnl));
    p0 += fast_exp2(__fmaf_rn(x2.x, LOG2E, -mnl));
    p1 += fast_exp2(__fmaf_rn(x2.y, LOG2E, -mnl));
    p2 += fast_exp2(__fmaf_rn(x2.z, LOG2E, -mnl));
    p3 += fast_exp2(__fmaf_rn(x2.w, LOG2E, -mnl));
    p0 += fast_exp2(__fmaf_rn(x3.x, LOG2E, -mnl));
    p1 += fast_exp2(__fmaf_rn(x3.y, LOG2E, -mnl));
    p2 += fast_exp2(__fmaf_rn(x3.z, LOG2E, -mnl));
    p3 += fast_exp2(__fmaf_rn(x3.w, LOG2E, -mnl));
    s += (p0 + p1) + (p2 + p3);
  }

  // Reduce (m, s) across the wave (wave32), then across the 8 waves via LDS.
#pragma unroll
  for (int off = 16; off > 0; off >>= 1) {
    float mo = __shfl_down(m, off, 32);
    float so = __shfl_down(s, off, 32);
    lse_merge(m, s, mo, so);
  }
  __shared__ float wm[THREADS / 32];
  __shared__ float wsum[THREADS / 32];
  const int wave = tid >> 5;
  if ((tid & 31) == 0) { wm[wave] = m; wsum[wave] = s; }
  __syncthreads();

  if (tid == 0) {
    float M = wm[0], S = wsum[0];
#pragma unroll
    for (int w = 1; w < THREADS / 32; ++w) lse_merge(M, S, wm[w], wsum[w]);

    const int base = b * TT + t + 1;          // shifted label/mask/step
    const float mk = (float)loss_mask[base];
    if (mk != 0.0f) {
      const int label = input_ids[base];
      const int st = step_indices[base];
      const float xlab = rowp[label];
      const float ce = M + fast_log2(S) * LN2 - xlab;  // -log_softmax at the label
      atomicAdd(&sums[b * SS + st], ce);
      atomicAdd(&cnts[b * SS + st], 1.0f);
    }
  }
}

__global__ void finalize_kernel(const float* __restrict__ sums,
                                const float* __restrict__ cnts,
                                float* __restrict__ out) {
  if (threadIdx.x != 0 || blockIdx.x != 0) return;
  float total = 0.0f, nsamp = 0.0f;
  for (int b = 0; b < BB; ++b) {
    float pos = 0.0f, tce = 0.0f, tcnt = 0.0f;
    bool has_pos = false;
    for (int s = 0; s < SS; ++s) {
      float c = cnts[b * SS + s];
      float sm = sums[b * SS + s];
      tce += sm;
      tcnt += c;
      if (s >= 1 && c > 0.0f) { pos += sm / c; has_pos = true; }
    }
    const bool active = tcnt > 0.0f;
    const float fallback = tce / fmaxf(tcnt, 1.0f);
    const float sl = has_pos ? pos : fallback;
    if (active) { total += sl; nsamp += 1.0f; }
  }
  out[0] = (nsamp > 0.0f) ? total / nsamp : 0.0f;
}

// -------------------------------------------------------------------------------------
extern "C" void kernel_launch(void* const* d_in, const int* in_sizes, int n_in,
                              void* d_out, int out_size, void* d_ws, size_t ws_size,
                              hipStream_t stream) {
  const float* logits = (const float*)d_in[0];
  const int* input_ids = (const int*)d_in[1];
  const int* loss_mask = (const int*)d_in[2];
  const int* step_indices = (const int*)d_in[3];

  float* ws = (float*)d_ws;          // [0..67] = sums, [68..135] = counts
  float* sums = ws;
  float* cnts = ws + BB * SS;

  zero_ws_kernel<<<1, 256, 0, stream>>>(ws, 2 * BB * SS);

  const size_t shmem = NBUF * CHUNK_V4 * sizeof(float4);  // 48 KB triple buffer
  row_ce_kernel<<<ROWS, THREADS, shmem, stream>>>(logits, input_ids, loss_mask,
                                                  step_indices, sums, cnts);

  finalize_kernel<<<1, 32, 0, stream>>>(sums, cnts, (float*)d_out);
}